// GNN_77850577207790
// MI455X (gfx1250) — compile-verified
//
#include <hip/hip_runtime.h>

typedef __attribute__((ext_vector_type(2))) float v2f;
typedef __attribute__((ext_vector_type(8))) float v8f;

#define IN_CH 256
#define HID_CH 32
#define OUT_CH 8
#define N_GRAPHS 64

// ---------------- degree / normalization ----------------
__global__ void k_fill_deg(float* __restrict__ deg, int n) {
    int i = blockIdx.x * blockDim.x + threadIdx.x;
    if (i < n) deg[i] = 1.0f;  // self-loop contribution
}

__global__ void k_deg_edges(const int* __restrict__ dst, float* __restrict__ deg, int ne) {
    int e = blockIdx.x * blockDim.x + threadIdx.x;
    if (e < ne) atomicAdd(&deg[dst[e]], 1.0f);
}

__global__ void k_dinv(float* __restrict__ deg, int n) {
    int i = blockIdx.x * blockDim.x + threadIdx.x;
    if (i < n) deg[i] = rsqrtf(deg[i]);  // deg >= 1 always (self-loops)
}

// ---------------- GEMM1: xw = x @ W1  [N,256]x[256,32], fp32 WMMA ----------------
__global__ void k_gemm1(const float* __restrict__ x, const float* __restrict__ W,
                        float* __restrict__ xw, int ntiles) {
    __shared__ float sW[IN_CH * HID_CH];  // 32 KB
    for (int i = threadIdx.x; i < IN_CH * HID_CH; i += blockDim.x) sW[i] = W[i];
    __syncthreads();

    const int lane = threadIdx.x & 31;
    const int wave = threadIdx.x >> 5;
    const int tile = blockIdx.x * 8 + wave;
    if (tile >= ntiles) return;           // wave-uniform: EXEC stays all-ones

    const int row0 = tile * 16;
    const int m    = lane & 15;           // M (for A) / N (for B,C)
    const int kh   = (lane >> 4) << 1;    // 0 or 2: K sub-offset per half-wave
    const float* xrow = x + (size_t)(row0 + m) * IN_CH + kh;

    v8f c0 = {}; v8f c1 = {};
#pragma unroll 4
    for (int k = 0; k < IN_CH; k += 4) {
        // A 16x4 fp32 layout: v0 = {K=k (lanes0-15), K=k+2 (lanes16-31)}, v1 = +1
        v2f a = *(const v2f*)(xrow + k);
        const float* wp = &sW[(k + kh) * HID_CH + m];
        v2f b0 = { wp[0],  wp[HID_CH] };           // cols 0..15
        v2f b1 = { wp[16], wp[HID_CH + 16] };      // cols 16..31
        c0 = __builtin_amdgcn_wmma_f32_16x16x4_f32(false, a, false, b0, (short)0, c0, false, false);
        c1 = __builtin_amdgcn_wmma_f32_16x16x4_f32(false, a, false, b1, (short)0, c1, false, false);
    }

    // C/D layout: VGPR r -> M=r (lanes 0-15) / M=r+8 (lanes 16-31), N = lane&15
    const int mb = (lane >> 4) << 3;
    float* orow = xw + (size_t)(row0 + mb) * HID_CH + m;
#pragma unroll
    for (int r = 0; r < 8; ++r) {
        orow[(size_t)r * HID_CH]      = c0[r];
        orow[(size_t)r * HID_CH + 16] = c1[r];
    }
}

// ---------------- GEMM2: xw2 = h1 @ W2  [N,32]x[32,8] (N padded to 16) ----------------
__global__ void k_gemm2(const float* __restrict__ h1, const float* __restrict__ W,
                        float* __restrict__ xw2, int ntiles) {
    __shared__ float sW[HID_CH * 16];     // 32x16, cols 8..15 zero-padded
    for (int i = threadIdx.x; i < HID_CH * 16; i += blockDim.x) {
        int r = i >> 4, c = i & 15;
        sW[i] = (c < OUT_CH) ? W[r * OUT_CH + c] : 0.0f;
    }
    __syncthreads();

    const int lane = threadIdx.x & 31;
    const int wave = threadIdx.x >> 5;
    const int tile = blockIdx.x * 8 + wave;
    if (tile >= ntiles) return;

    const int row0 = tile * 16;
    const int m    = lane & 15;
    const int kh   = (lane >> 4) << 1;
    const float* arow = h1 + (size_t)(row0 + m) * HID_CH + kh;

    v8f c0 = {};
#pragma unroll
    for (int k = 0; k < HID_CH; k += 4) {
        v2f a = *(const v2f*)(arow + k);
        const float* wp = &sW[(k + kh) * 16 + m];
        v2f b = { wp[0], wp[16] };
        c0 = __builtin_amdgcn_wmma_f32_16x16x4_f32(false, a, false, b, (short)0, c0, false, false);
    }

    const int mb = (lane >> 4) << 3;
    if (m < OUT_CH) {                      // only the 8 valid columns
        float* orow = xw2 + (size_t)(row0 + mb) * OUT_CH + m;
#pragma unroll
        for (int r = 0; r < 8; ++r) orow[(size_t)r * OUT_CH] = c0[r];
    }
}

// ---------------- self-loop + bias init:  h = xw * dinv^2 + b ----------------
__global__ void k_self_init(const float* __restrict__ xw, const float* __restrict__ dinv,
                            const float* __restrict__ b, float* __restrict__ h,
                            int total, int shift, int mask) {
    int i = blockIdx.x * blockDim.x + threadIdx.x;
    if (i < total) {
        int node = i >> shift;
        float di = dinv[node];
        h[i] = xw[i] * di * di + b[i & mask];
    }
}

// ---------------- edge scatter, 32 channels: one wave per edge ----------------
__global__ void k_scatter32(const int* __restrict__ src, const int* __restrict__ dst,
                            const float* __restrict__ dinv, const float* __restrict__ xw,
                            float* __restrict__ h, int ne) {
    int gid  = blockIdx.x * blockDim.x + threadIdx.x;
    int e    = gid >> 5;
    int lane = gid & 31;
    if (e < ne) {
        int s = src[e], d = dst[e];
        float w = dinv[s] * dinv[d];
        atomicAdd(&h[(size_t)d * HID_CH + lane], xw[(size_t)s * HID_CH + lane] * w);
    }
}

// ---------------- edge scatter, 8 channels: 1/4 wave per edge ----------------
__global__ void k_scatter8(const int* __restrict__ src, const int* __restrict__ dst,
                           const float* __restrict__ dinv, const float* __restrict__ xw,
                           float* __restrict__ h, int ne) {
    int gid = blockIdx.x * blockDim.x + threadIdx.x;
    int e   = gid >> 3;
    int c   = gid & 7;
    if (e < ne) {
        int s = src[e], d = dst[e];
        float w = dinv[s] * dinv[d];
        atomicAdd(&h[(size_t)d * OUT_CH + c], xw[(size_t)s * OUT_CH + c] * w);
    }
}

__global__ void k_relu(float* __restrict__ h, int total) {
    int i = blockIdx.x * blockDim.x + threadIdx.x;
    if (i < total) h[i] = fmaxf(h[i], 0.0f);
}

// ---------------- pooling: LDS-hierarchical segment sum ----------------
__global__ void k_zero(float* __restrict__ p, int n) {
    int i = blockIdx.x * blockDim.x + threadIdx.x;
    if (i < n) p[i] = 0.0f;
}

__global__ void k_pool(const int* __restrict__ batch, const float* __restrict__ h2,
                       float* __restrict__ pool, float* __restrict__ cnt, int n) {
    __shared__ float sp[N_GRAPHS * OUT_CH];
    __shared__ float sc[N_GRAPHS];
    for (int i = threadIdx.x; i < N_GRAPHS * OUT_CH; i += blockDim.x) sp[i] = 0.0f;
    for (int i = threadIdx.x; i < N_GRAPHS; i += blockDim.x) sc[i] = 0.0f;
    __syncthreads();
    int node = blockIdx.x * blockDim.x + threadIdx.x;
    if (node < n) {
        int g = batch[node];
        atomicAdd(&sc[g], 1.0f);
#pragma unroll
        for (int c = 0; c < OUT_CH; ++c)
            atomicAdd(&sp[g * OUT_CH + c], h2[(size_t)node * OUT_CH + c]);
    }
    __syncthreads();
    for (int i = threadIdx.x; i < N_GRAPHS * OUT_CH; i += blockDim.x)
        atomicAdd(&pool[i], sp[i]);
    for (int i = threadIdx.x; i < N_GRAPHS; i += blockDim.x)
        atomicAdd(&cnt[i], sc[i]);
}

// ---------------- mean + log_softmax, [64,8] ----------------
__global__ void k_final(const float* __restrict__ pool, const float* __restrict__ cnt,
                        float* __restrict__ out) {
    int g = threadIdx.x;
    if (g < N_GRAPHS) {
        float c = fmaxf(cnt[g], 1.0f);
        float v[OUT_CH];
        float mx = -1e30f;
#pragma unroll
        for (int i = 0; i < OUT_CH; ++i) {
            v[i] = pool[g * OUT_CH + i] / c;
            mx = fmaxf(mx, v[i]);
        }
        float s = 0.0f;
#pragma unroll
        for (int i = 0; i < OUT_CH; ++i) s += expf(v[i] - mx);
        float lse = logf(s);
#pragma unroll
        for (int i = 0; i < OUT_CH; ++i) out[g * OUT_CH + i] = v[i] - mx - lse;
    }
}

extern "C" void kernel_launch(void* const* d_in, const int* in_sizes, int n_in,
                              void* d_out, int out_size, void* d_ws, size_t ws_size,
                              hipStream_t stream) {
    const float* x     = (const float*)d_in[0];
    const int*   ei    = (const int*)  d_in[1];
    const int*   batch = (const int*)  d_in[2];
    const float* W1    = (const float*)d_in[3];
    const float* b1    = (const float*)d_in[4];
    const float* W2    = (const float*)d_in[5];
    const float* b2    = (const float*)d_in[6];
    float* out = (float*)d_out;

    const int nn = in_sizes[0] / IN_CH;  // 100000 nodes
    const int ne = in_sizes[1] / 2;      // 1600000 edges
    const int* src = ei;
    const int* dst = ei + ne;

    // ---- carve workspace (~32.4 MB) ----
    char* ws = (char*)d_ws;
    size_t off = 0;
    auto carve = [&](size_t bytes) {
        char* p = ws + off;
        off = (off + bytes + 255) & ~(size_t)255;
        return (float*)p;
    };
    float* dinv = carve((size_t)nn * sizeof(float));
    float* xw   = carve((size_t)nn * HID_CH * sizeof(float));
    float* h1   = carve((size_t)nn * HID_CH * sizeof(float));
    float* xw2  = carve((size_t)nn * OUT_CH * sizeof(float));
    float* h2   = carve((size_t)nn * OUT_CH * sizeof(float));
    float* pool = carve((size_t)(N_GRAPHS * OUT_CH + N_GRAPHS) * sizeof(float));
    float* cnt  = pool + N_GRAPHS * OUT_CH;
    (void)ws_size; (void)n_in; (void)out_size;

    const int B = 256;
    const int ntiles = (nn + 15) / 16;

    // degree / dinv
    k_fill_deg <<<(nn + B - 1) / B, B, 0, stream>>>(dinv, nn);
    k_deg_edges<<<(ne + B - 1) / B, B, 0, stream>>>(dst, dinv, ne);
    k_dinv     <<<(nn + B - 1) / B, B, 0, stream>>>(dinv, nn);

    // layer 1
    k_gemm1<<<(ntiles + 7) / 8, B, 0, stream>>>(x, W1, xw, ntiles);
    {
        int tot = nn * HID_CH;
        k_self_init<<<(tot + B - 1) / B, B, 0, stream>>>(xw, dinv, b1, h1, tot, 5, 31);
        long long thr = (long long)ne * 32;
        k_scatter32<<<(int)((thr + B - 1) / B), B, 0, stream>>>(src, dst, dinv, xw, h1, ne);
        k_relu<<<(tot + B - 1) / B, B, 0, stream>>>(h1, tot);
    }

    // layer 2
    k_gemm2<<<(ntiles + 7) / 8, B, 0, stream>>>(h1, W2, xw2, ntiles);
    {
        int tot = nn * OUT_CH;
        k_self_init<<<(tot + B - 1) / B, B, 0, stream>>>(xw2, dinv, b2, h2, tot, 3, 7);
        long long thr = (long long)ne * 8;
        k_scatter8<<<(int)((thr + B - 1) / B), B, 0, stream>>>(src, dst, dinv, xw2, h2, ne);
        k_relu<<<(tot + B - 1) / B, B, 0, stream>>>(h2, tot);
    }

    // pooling + log_softmax
    k_zero <<<1, B, 0, stream>>>(pool, N_GRAPHS * OUT_CH + N_GRAPHS);
    k_pool <<<(nn + B - 1) / B, B, 0, stream>>>(batch, h2, pool, cnt, nn);
    k_final<<<1, 64, 0, stream>>>(pool, cnt, out);
}